// ModeTransformer_28226525069840
// MI455X (gfx1250) — compile-verified
//
#include <hip/hip_runtime.h>

typedef float v2f __attribute__((ext_vector_type(2)));
typedef float v8f __attribute__((ext_vector_type(8)));
typedef int   v4i __attribute__((ext_vector_type(4)));

#define HDIM 64
#define WDIM 64
#define TDIM 64
#define CDIM 96
#define NHEAD 6
#define HD 16
#define KTROWS 18        // t0-1 .. t0+16
#define SCSTR 28         // padded score row stride
#define TPB 192          // 6 waves (wave32), one head per wave

// ---- CDNA5 async global->LDS path (ASYNCcnt), with sync fallback ----
#if __has_builtin(__builtin_amdgcn_global_load_async_to_lds_b128) && \
    __has_builtin(__builtin_amdgcn_global_load_async_to_lds_b32)  && \
    __has_builtin(__builtin_amdgcn_s_wait_asynccnt)
#define USE_ASYNC 1
#else
#define USE_ASYNC 0
#endif

typedef __attribute__((address_space(1))) v4i* gv4p;   // global int4*
typedef __attribute__((address_space(3))) v4i* lv4p;   // LDS int4*
typedef __attribute__((address_space(1))) int* gi32p;  // global int*
typedef __attribute__((address_space(3))) int* li32p;  // LDS int*

__device__ __forceinline__ void cp16_g2l(const float* g, float* l) {
#if USE_ASYNC
    __builtin_amdgcn_global_load_async_to_lds_b128((gv4p)g, (lv4p)l, 0, 0);
#else
    *(float4*)l = *(const float4*)g;
#endif
}
__device__ __forceinline__ void cp4_g2l(const float* g, float* l) {
#if USE_ASYNC
    __builtin_amdgcn_global_load_async_to_lds_b32((gi32p)g, (li32p)l, 0, 0);
#else
    *l = *g;
#endif
}
__device__ __forceinline__ void wait_g2l() {
#if USE_ASYNC
    __builtin_amdgcn_s_wait_asynccnt(0);
#endif
}

__global__ __launch_bounds__(TPB)
void natten3d_wmma_kernel(const float* __restrict__ qg,
                          const float* __restrict__ kg,
                          const float* __restrict__ rpbg,
                          float* __restrict__ outg)
{
    __shared__ __align__(16) float ktile[9 * KTROWS * CDIM];   // 15552: [ii*3+jj][trow][c]
    __shared__ __align__(16) float qtile[16 * CDIM];           // 1536: [t][c], UNscaled
    __shared__ float sc[NHEAD * 16 * SCSTR];                   // per-head score rows
    __shared__ float rpbs[NHEAD * 27];

    const int bx = blockIdx.x;
    const int tt = bx & 3;
    const int w  = (bx >> 2) & 63;
    const int h  = (bx >> 8) & 63;
    const int b  = bx >> 14;
    const int t0 = tt << 4;

    const int tid = threadIdx.x;

    // ---------------- stage k neighborhood in 16B chunks (zero padded) ----------------
    // 9*18 rows of 96 floats = 162 rows * 24 chunks = 3888 chunks; bounds are per-row.
    for (int f4 = tid; f4 < 9 * KTROWS * (CDIM / 4); f4 += TPB) {
        const int c4   = f4 % (CDIM / 4);
        const int row  = f4 / (CDIM / 4);
        const int trow = row % KTROWS;
        const int sp   = row / KTROWS;        // ii*3+jj
        const int jj   = sp % 3;
        const int ii   = sp / 3;
        const int hh = h + ii - 1;
        const int ww = w + jj - 1;
        const int t  = t0 + trow - 1;
        float* l = &ktile[f4 * 4];
        if ((unsigned)hh < 64u && (unsigned)ww < 64u && (unsigned)t < 64u) {
            const int gidx = ((b * HDIM + hh) * WDIM + ww) * (TDIM * CDIM) + t * CDIM + c4 * 4;
            cp16_g2l(&kg[gidx], l);
        } else {
            *(float4*)l = make_float4(0.f, 0.f, 0.f, 0.f);
        }
    }
    // ---------------- stage q tile: 16 contiguous rows = 1536 floats ----------------
    {
        const int qbase = ((b * HDIM + h) * WDIM + w) * (TDIM * CDIM) + t0 * CDIM;
        for (int f4 = tid; f4 < 16 * (CDIM / 4); f4 += TPB)
            cp16_g2l(&qg[qbase + f4 * 4], &qtile[f4 * 4]);
    }
    if (tid < NHEAD * 27) cp4_g2l(&rpbg[tid], &rpbs[tid]);

    wait_g2l();
    __syncthreads();

    const int wave = tid >> 5;     // head index (0..5)
    const int lane = tid & 31;
    const int n    = wave;
    const int M    = lane & 15;    // also column index N for B/D fragments
    const int half = lane >> 4;    // 0 or 1

    // ---------------- preload A fragments (Q tile, scaled by HD^-0.5 = 0.25) ----------------
    // A 16x4 fp32 layout: lanes 0-15 hold K={0,1} in 2 VGPRs, lanes 16-31 hold K={2,3}
    float aA[4][2];
#pragma unroll
    for (int kk = 0; kk < 4; ++kk) {
#pragma unroll
        for (int v = 0; v < 2; ++v)
            aA[kk][v] = qtile[M * CDIM + n * HD + kk * 4 + 2 * half + v] * 0.25f;
    }

    // ---------------- 9 banded GEMMs: D = Q(16x16) x K^T(16x16) per (ii,jj) ----------------
    for (int ij = 0; ij < 9; ++ij) {
        v8f acc = {};
#pragma unroll
        for (int kk = 0; kk < 4; ++kk) {
            v2f a, bb;
            a.x = aA[kk][0];
            a.y = aA[kk][1];
            // B 4x16 fp32: row K striped across lanes; lanes0-15 K={0,1}, lanes16-31 K={2,3}
            // column N = lane&15 -> key t-row = N+1 (ktile row 0 is t0-1)
            const int kidx = (ij * KTROWS + (M + 1)) * CDIM + n * HD + kk * 4 + 2 * half;
            bb.x = ktile[kidx + 0];
            bb.y = ktile[kidx + 1];
            acc = __builtin_amdgcn_wmma_f32_16x16x4_f32(
                false, a, false, bb, (short)0, acc, false, false);
        }
        // extract band entries: D vgpr r holds (M = r + 8*half, N = lane&15);
        // needed when l = N - M + 1 in {0,1,2}
#pragma unroll
        for (int r = 0; r < 8; ++r) {
            const int Mr   = r + 8 * half;
            const int diff = M - Mr;           // N - Mr
            if (diff >= -1 && diff <= 1)
                sc[(n * 16 + Mr) * SCSTR + ij * 3 + (diff + 1)] = acc[r];
        }
    }

    // ---------------- corner patches: (t=0,l=0,key t0-1) and (t=15,l=2,key t0+16) ----------------
    if (lane < 9) {
        const int ij = lane;
        float s = 0.0f;
#pragma unroll
        for (int d = 0; d < HD; ++d)
            s += qtile[0 * CDIM + n * HD + d] * ktile[(ij * KTROWS + 0) * CDIM + n * HD + d];
        sc[(n * 16 + 0) * SCSTR + ij * 3 + 0] = s * 0.25f;
    }
    if (lane >= 16 && lane < 25) {
        const int ij = lane - 16;
        float s = 0.0f;
#pragma unroll
        for (int d = 0; d < HD; ++d)
            s += qtile[15 * CDIM + n * HD + d] * ktile[(ij * KTROWS + 17) * CDIM + n * HD + d];
        sc[(n * 16 + 15) * SCSTR + ij * 3 + 2] = s * 0.25f;
    }

    // ---------------- bias + softmax + offset contraction, lanes 0..15 own one t each ----------------
    if (lane < 16) {
        const int t = lane;
        const float* srow = &sc[(n * 16 + t) * SCSTR];
        const float* rrow = &rpbs[n * 27];
        float v[27];
        float mx = -1e30f;
#pragma unroll
        for (int idx = 0; idx < 27; ++idx) {
            v[idx] = srow[idx] + rrow[idx];
            mx = fmaxf(mx, v[idx]);
        }
        float s = 0.0f, sx = 0.0f, sy = 0.0f, sz = 0.0f;
#pragma unroll
        for (int idx = 0; idx < 27; ++idx) {
            const float e = __expf(v[idx] - mx);
            s  += e;
            sx += e * (float)(idx / 9 - 1);
            sy += e * (float)((idx / 3) % 3 - 1);
            sz += e * (float)(idx % 3 - 1);
        }
        const float inv = 1.0f / s;
        const int cs    = TDIM * WDIM * HDIM;   // channel stride 262144
        const int obase = ((b * (NHEAD * 3) + n * 3) * HDIM + h) * (WDIM * TDIM) + w * TDIM + t0 + t;
        outg[obase]          = sx * inv;
        outg[obase + cs]     = sy * inv;
        outg[obase + 2 * cs] = sz * inv;
    }
}

extern "C" void kernel_launch(void* const* d_in, const int* in_sizes, int n_in,
                              void* d_out, int out_size, void* d_ws, size_t ws_size,
                              hipStream_t stream) {
    const float* q   = (const float*)d_in[0];
    const float* k   = (const float*)d_in[1];
    const float* rpb = (const float*)d_in[2];
    float* out = (float*)d_out;
    // blocks: b(2) * h(64) * w(64) * t-tiles(4) = 32768
    natten3d_wmma_kernel<<<dim3(2 * 64 * 64 * 4), dim3(TPB), 0, stream>>>(q, k, rpb, out);
}